// DetectHead_multispectral_19061064860264
// MI455X (gfx1250) — compile-verified
//
#include <hip/hip_runtime.h>
#include <stdint.h>

#define NANCH     34128
#define NMOD_ANCH 17064
#define NB        8
#define NCLS      80
#define KTOP      1000
#define KPAD      1024
#define SCORE_THR 0.05f
#define NMS_THR   0.6f

typedef float v2f __attribute__((ext_vector_type(2)));
typedef float v8f __attribute__((ext_vector_type(8)));
typedef int v4i_ __attribute__((vector_size(4 * sizeof(int))));  // type expected by async builtin

#if defined(__AMDGCN__) && __has_builtin(__builtin_amdgcn_wmma_f32_16x16x4_f32)
#define HAVE_WMMA_F32_16X16X4 1
#else
#define HAVE_WMMA_F32_16X16X4 0
#endif

#if defined(__AMDGCN__) && __has_builtin(__builtin_amdgcn_global_load_async_to_lds_b128)
#define HAVE_ASYNC_LDS 1
#else
#define HAVE_ASYNC_LDS 0
#endif

struct Inputs {
  const float* cls[2][5];
  const float* cnt[2][5];
  const float* reg[2][5];
};

__device__ __forceinline__ float sigmoidf_(float x) { return 1.0f / (1.0f + __expf(-x)); }

__device__ __forceinline__ void asyncCopyB128(const float* gsrc, float* ldsDst) {
#if HAVE_ASYNC_LDS
  __attribute__((address_space(1))) v4i_* g =
      (__attribute__((address_space(1))) v4i_*)(uintptr_t)gsrc;
  __attribute__((address_space(3))) v4i_* l =
      (__attribute__((address_space(3))) v4i_*)(uint32_t)(uintptr_t)ldsDst;
  __builtin_amdgcn_global_load_async_to_lds_b128(g, l, 0, 0);
#else
  float4 t = *(const float4*)gsrc;
  *(float4*)ldsDst = t;
#endif
}

__device__ __forceinline__ void waitAsync() {
#if defined(__AMDGCN__)
#if __has_builtin(__builtin_amdgcn_s_wait_asynccnt)
  __builtin_amdgcn_s_wait_asynccnt(0);
#else
  asm volatile("s_wait_asynccnt 0" ::: "memory");
#endif
#endif
}

// ---------------- Kernel 1: fused per-anchor score / class / box ----------------
__global__ void score_box_kernel(Inputs in, float* __restrict__ scores,
                                 int* __restrict__ classes, float* __restrict__ boxes) {
  int anchor = blockIdx.x * blockDim.x + threadIdx.x;
  if (anchor >= NANCH) return;
  int b = blockIdx.y;
  int mod = (anchor >= NMOD_ANCH) ? 1 : 0;
  int a = anchor - mod * NMOD_ANCH;
  int li, off, w, s, hw;
  if (a < 12800)      { li = 0; off = 0;     w = 128; s = 8;   hw = 12800; }
  else if (a < 16000) { li = 1; off = 12800; w = 64;  s = 16;  hw = 3200; }
  else if (a < 16800) { li = 2; off = 16000; w = 32;  s = 32;  hw = 800; }
  else if (a < 17008) { li = 3; off = 16800; w = 16;  s = 64;  hw = 208; }
  else                { li = 4; off = 17008; w = 8;   s = 128; hw = 56; }
  int pos = a - off;
  int y = pos / w, x = pos - y * w;
  float cx = (float)(x * s + (s >> 1));
  float cy = (float)(y * s + (s >> 1));

  const float* cp = in.cls[mod][li] + (size_t)b * NCLS * hw + pos;
  float m = cp[0];
  int am = 0;
  for (int c = 1; c < NCLS; ++c) {  // coalesced: lanes hit consecutive pos per channel
    float v = cp[(size_t)c * hw];
    if (v > m) { m = v; am = c; }
  }
  float pc = sigmoidf_(m);  // max(sigmoid(x)) == sigmoid(max(x)) (monotone)
  float pt = sigmoidf_(in.cnt[mod][li][(size_t)b * hw + pos]);
  float sc = sqrtf(pc * pt);

  const float* rp = in.reg[mod][li] + (size_t)b * 4 * hw + pos;
  float r0 = rp[0], r1 = rp[(size_t)hw], r2 = rp[(size_t)2 * hw], r3 = rp[(size_t)3 * hw];

  size_t gi = (size_t)b * NANCH + anchor;
  scores[gi] = sc;
  classes[gi] = am + 1;
  boxes[gi * 4 + 0] = cx - r0;
  boxes[gi * 4 + 1] = cy - r1;
  boxes[gi * 4 + 2] = cx + r2;
  boxes[gi * 4 + 3] = cy + r3;
}

// ---------------- Kernel 2: per-batch radix-select top-1000 + bitonic sort ----------------
__global__ void topk_kernel(const float* __restrict__ scores, const int* __restrict__ classes,
                            const float* __restrict__ boxes, float* __restrict__ tk_score,
                            int* __restrict__ tk_cls, float* __restrict__ tk_box) {
  const int b = blockIdx.x;
  const unsigned tid = threadIdx.x;  // 1024
  const float* sc = scores + (size_t)b * NANCH;

  __shared__ unsigned hist[256];
  __shared__ unsigned skey[KPAD];
  __shared__ unsigned sidx[KPAD];
  __shared__ unsigned sh_prefix, sh_want, sh_cntGT, sh_cntEQ;

  if (tid == 0) { sh_prefix = 0u; sh_want = KTOP; }

  // MSB-first radix select over positive-float bit patterns (scores > 0 -> order preserved)
  for (int pass = 0; pass < 4; ++pass) {
    if (tid < 256) hist[tid] = 0u;
    __syncthreads();
    unsigned prefix = sh_prefix, want = sh_want;
    int shift = 24 - 8 * pass;
    unsigned himask = (pass == 0) ? 0u : (0xFFFFFFFFu << (shift + 8));
    for (int i = tid; i < NANCH; i += 1024) {
      unsigned key = __float_as_uint(sc[i]);
      if ((key & himask) == prefix) atomicAdd(&hist[(key >> shift) & 255u], 1u);
    }
    __syncthreads();
    if (tid == 0) {
      unsigned cum = 0;
      int d = 255;
      for (; d > 0; --d) {
        unsigned c = hist[d];
        if (cum + c >= want) break;
        cum += c;
      }
      sh_prefix = prefix | ((unsigned)d << shift);
      sh_want = want - cum;
    }
    __syncthreads();
  }
  const unsigned T = sh_prefix;
  const unsigned wantEq = sh_want;

  if (tid == 0) { sh_cntGT = 0u; sh_cntEQ = 0u; }
  __syncthreads();
  for (int i = tid; i < NANCH; i += 1024) {
    unsigned key = __float_as_uint(sc[i]);
    if (key > T) {
      unsigned p = atomicAdd(&sh_cntGT, 1u);
      skey[p] = key;
      sidx[p] = (unsigned)i;
    }
  }
  __syncthreads();
  const unsigned G = sh_cntGT;
  for (int i = tid; i < NANCH; i += 1024) {
    unsigned key = __float_as_uint(sc[i]);
    if (key == T) {
      unsigned q = atomicAdd(&sh_cntEQ, 1u);
      if (q < wantEq) { skey[G + q] = key; sidx[G + q] = (unsigned)i; }
    }
  }
  __syncthreads();
  if (tid >= KTOP) { skey[tid] = 0u; sidx[tid] = 0u; }

  // bitonic sort, descending by key
  for (unsigned k2 = 2; k2 <= KPAD; k2 <<= 1) {
    for (unsigned j = k2 >> 1; j > 0; j >>= 1) {
      __syncthreads();
      unsigned ixj = tid ^ j;
      if (ixj > tid) {
        bool up = ((tid & k2) == 0);
        unsigned ka = skey[tid], kb = skey[ixj];
        bool sw = up ? (ka < kb) : (ka > kb);
        if (sw) {
          skey[tid] = kb; skey[ixj] = ka;
          unsigned ia = sidx[tid]; sidx[tid] = sidx[ixj]; sidx[ixj] = ia;
        }
      }
    }
  }
  __syncthreads();

  size_t gi = (size_t)b * KPAD + tid;
  unsigned src = sidx[tid];
  tk_score[gi] = __uint_as_float(skey[tid]);
  tk_cls[gi] = classes[(size_t)b * NANCH + src];
  size_t sb = ((size_t)b * NANCH + src) * 4;
  tk_box[gi * 4 + 0] = boxes[sb + 0];
  tk_box[gi * 4 + 1] = boxes[sb + 1];
  tk_box[gi * 4 + 2] = boxes[sb + 2];
  tk_box[gi * 4 + 3] = boxes[sb + 3];
}

// ---------------- Kernel 3: NMS prep (max_coord reduce, class offsets, areas) ----------------
__global__ void nms_prep_kernel(const float* __restrict__ tk_score, const int* __restrict__ tk_cls,
                                const float* __restrict__ tk_box, float* __restrict__ nms_box,
                                float* __restrict__ area, unsigned* __restrict__ validf) {
  int b = blockIdx.x, t = threadIdx.x;  // 1024
  __shared__ float red[1024];
  size_t gi = (size_t)b * KPAD + t;
  float sc = tk_score[gi];
  float x1 = tk_box[gi * 4 + 0], y1 = tk_box[gi * 4 + 1];
  float x2 = tk_box[gi * 4 + 2], y2 = tk_box[gi * 4 + 3];
  bool valid = (t < KTOP) && (sc >= SCORE_THR);
  red[t] = valid ? fmaxf(fmaxf(x1, y1), fmaxf(x2, y2)) : -__builtin_inff();
  __syncthreads();
  for (int st = 512; st > 0; st >>= 1) {
    if (t < st) red[t] = fmaxf(red[t], red[t + st]);
    __syncthreads();
  }
  float maxc = red[0];
  if (!(maxc > -3.0e38f)) maxc = 0.0f;
  float offv = (float)tk_cls[gi] * (maxc + 1.0f);
  float nx1 = x1 + offv, ny1 = y1 + offv, nx2 = x2 + offv, ny2 = y2 + offv;
  nms_box[gi * 4 + 0] = nx1;
  nms_box[gi * 4 + 1] = ny1;
  nms_box[gi * 4 + 2] = nx2;
  nms_box[gi * 4 + 3] = ny2;
  area[gi] = (nx2 - nx1 + 1.0f) * (ny2 - ny1 + 1.0f);
  validf[gi] = valid ? 1u : 0u;
}

// ---------------- Kernel 4: suppression bitmask; wave per 16x32 supertile ----------------
__global__ void iou_mask_kernel(const float* __restrict__ nms_box, const float* __restrict__ area,
                                unsigned* __restrict__ mask) {
  int b = blockIdx.x;
  int tid = threadIdx.x;  // 1024 = 32 waves
  __shared__ float sbox[KPAD * 4];  // 16 KB
  __shared__ float sarea[KPAD];     // 4 KB

  const float* gb = nms_box + (size_t)b * KPAD * 4;
  const float* ga = area + (size_t)b * KPAD;
  asyncCopyB128(gb + tid * 4, &sbox[tid * 4]);                  // 1024 x 16B
  if (tid < 256) asyncCopyB128(ga + tid * 4, &sarea[tid * 4]);  // 256 x 16B
  waitAsync();
  __syncthreads();

  const int lane = tid & 31;
  const int wave = tid >> 5;
  const bool lo = lane < 16;

  for (int st = wave; st < 2048; st += 32) {  // 64 row-tiles x 32 col-supertiles
    int tr = st >> 5;   // 16-row tile
    int scs = st & 31;  // 32-col supertile -> mask word index
    int c0 = scs * 32, c1 = scs * 32 + 16;

    // WMMA operands: A(16x4) col0=area_row,col1=1 ; B(4x16) row0=1,row1=area_col
    v2f a2, b0v, b1v;
    a2.x = lo ? sarea[tr * 16 + lane] : 0.0f;
    a2.y = lo ? 1.0f : 0.0f;
    b0v.x = lo ? 1.0f : 0.0f;
    b0v.y = lo ? sarea[c0 + lane] : 0.0f;
    b1v.x = lo ? 1.0f : 0.0f;
    b1v.y = lo ? sarea[c1 + lane] : 0.0f;
#if HAVE_WMMA_F32_16X16X4
    v8f cz = {0.f, 0.f, 0.f, 0.f, 0.f, 0.f, 0.f, 0.f};
    // asum[r] @ lane -> area_row[r + 8*(lane>>4)] + area_col[lane&15]
    v8f as0 = __builtin_amdgcn_wmma_f32_16x16x4_f32(false, a2, false, b0v, (short)0, cz, false, false);
    v8f as1 = __builtin_amdgcn_wmma_f32_16x16x4_f32(false, a2, false, b1v, (short)0, cz, false, false);
#else
    (void)a2; (void)b0v; (void)b1v;
#endif
    unsigned words[16];
    int jl = lane & 15;
    for (int r = 0; r < 8; ++r) {
      int i_loc = r + ((lane >> 4) << 3);
      int i = tr * 16 + i_loc;
      int j0 = c0 + jl, j1 = c1 + jl;
      float ix1 = sbox[i * 4 + 0], iy1 = sbox[i * 4 + 1];
      float ix2 = sbox[i * 4 + 2], iy2 = sbox[i * 4 + 3];
      float in0 = fmaxf(fminf(ix2, sbox[j0 * 4 + 2]) - fmaxf(ix1, sbox[j0 * 4 + 0]) + 1.0f, 0.0f) *
                  fmaxf(fminf(iy2, sbox[j0 * 4 + 3]) - fmaxf(iy1, sbox[j0 * 4 + 1]) + 1.0f, 0.0f);
      float in1 = fmaxf(fminf(ix2, sbox[j1 * 4 + 2]) - fmaxf(ix1, sbox[j1 * 4 + 0]) + 1.0f, 0.0f) *
                  fmaxf(fminf(iy2, sbox[j1 * 4 + 3]) - fmaxf(iy1, sbox[j1 * 4 + 1]) + 1.0f, 0.0f);
#if HAVE_WMMA_F32_16X16X4
      float s0 = as0[r], s1 = as1[r];
#else
      float s0 = sarea[i] + sarea[j0];
      float s1 = sarea[i] + sarea[j1];
#endif
      bool p0 = in0 > NMS_THR * (s0 - in0);  // iou > thr  (union > 0 always here)
      bool p1 = in1 > NMS_THR * (s1 - in1);
      unsigned bal0 = (unsigned)__ballot(p0);
      unsigned bal1 = (unsigned)__ballot(p1);
      words[r] = (bal0 & 0xFFFFu) | ((bal1 & 0xFFFFu) << 16);
      words[r + 8] = (bal0 >> 16) | ((bal1 >> 16) << 16);
    }
    if (lane == 0) {
      for (int q = 0; q < 16; ++q)
        mask[((size_t)b * KPAD + tr * 16 + q) * 32 + scs] = words[q];
    }
  }
}

// ---------------- Kernel 5: greedy scan on one wave + output write ----------------
__global__ void nms_scan_kernel(const float* __restrict__ tk_score, const int* __restrict__ tk_cls,
                                const float* __restrict__ tk_box, const unsigned* __restrict__ validf,
                                const unsigned* __restrict__ mask, float* __restrict__ out_sc,
                                float* __restrict__ out_cl, float* __restrict__ out_bx,
                                float* __restrict__ out_keep) {
  int b = blockIdx.x, tid = threadIdx.x;  // 1024
  __shared__ unsigned supp[32];
  if (tid < 32) supp[tid] = 0u;
  __syncthreads();
  for (int i = 0; i < KTOP; ++i) {
    if (tid < 32) {
      int w = i >> 5, bt = i & 31;
      bool keep_i = (validf[(size_t)b * KPAD + i] != 0u) && (((supp[w] >> bt) & 1u) == 0u);
      if (keep_i) {
        unsigned m = mask[((size_t)b * KPAD + i) * 32 + tid];
        if (tid < w) m = 0u;
        else if (tid == w) m &= (bt == 31) ? 0u : (0xFFFFFFFFu << (bt + 1));
        supp[tid] |= m;  // suppress only j > i
      }
    }
    __syncthreads();
  }
  if (tid < KTOP) {
    size_t gi = (size_t)b * KPAD + tid;
    bool keep = (validf[gi] != 0u) && (((supp[tid >> 5] >> (tid & 31)) & 1u) == 0u);
    size_t oi = (size_t)b * KTOP + tid;
    out_sc[oi] = keep ? tk_score[gi] : 0.0f;
    out_cl[oi] = keep ? (float)tk_cls[gi] : 0.0f;
    for (int k2 = 0; k2 < 4; ++k2) out_bx[oi * 4 + k2] = keep ? tk_box[gi * 4 + k2] : 0.0f;
    out_keep[oi] = keep ? 1.0f : 0.0f;
  }
}

// ---------------- workspace layout ----------------
static constexpr size_t OFF_SCORES = 0;
static constexpr size_t OFF_CLASSES = OFF_SCORES + (size_t)NB * NANCH * 4;
static constexpr size_t OFF_BOXES = OFF_CLASSES + (size_t)NB * NANCH * 4;
static constexpr size_t OFF_TKS = OFF_BOXES + (size_t)NB * NANCH * 16;
static constexpr size_t OFF_TKC = OFF_TKS + (size_t)NB * KPAD * 4;
static constexpr size_t OFF_TKB = OFF_TKC + (size_t)NB * KPAD * 4;
static constexpr size_t OFF_NMSB = OFF_TKB + (size_t)NB * KPAD * 16;
static constexpr size_t OFF_AREA = OFF_NMSB + (size_t)NB * KPAD * 16;
static constexpr size_t OFF_VALID = OFF_AREA + (size_t)NB * KPAD * 4;
static constexpr size_t OFF_MASK = OFF_VALID + (size_t)NB * KPAD * 4;

extern "C" void kernel_launch(void* const* d_in, const int* in_sizes, int n_in,
                              void* d_out, int out_size, void* d_ws, size_t ws_size,
                              hipStream_t stream) {
  (void)in_sizes; (void)n_in; (void)out_size; (void)ws_size;
  Inputs in;
  for (int mod = 0; mod < 2; ++mod)
    for (int li = 0; li < 5; ++li) {
      in.cls[mod][li] = (const float*)d_in[mod * 15 + li * 3 + 0];
      in.cnt[mod][li] = (const float*)d_in[mod * 15 + li * 3 + 1];
      in.reg[mod][li] = (const float*)d_in[mod * 15 + li * 3 + 2];
    }
  char* ws = (char*)d_ws;
  float* scores = (float*)(ws + OFF_SCORES);
  int* classes = (int*)(ws + OFF_CLASSES);
  float* boxes = (float*)(ws + OFF_BOXES);
  float* tk_s = (float*)(ws + OFF_TKS);
  int* tk_c = (int*)(ws + OFF_TKC);
  float* tk_b = (float*)(ws + OFF_TKB);
  float* nms_b = (float*)(ws + OFF_NMSB);
  float* areaw = (float*)(ws + OFF_AREA);
  unsigned* validf = (unsigned*)(ws + OFF_VALID);
  unsigned* maskw = (unsigned*)(ws + OFF_MASK);

  float* out_sc = (float*)d_out;
  float* out_cl = out_sc + (size_t)NB * KTOP;
  float* out_bx = out_cl + (size_t)NB * KTOP;
  float* out_keep = out_bx + (size_t)NB * KTOP * 4;

  dim3 gA((NANCH + 255) / 256, NB);
  score_box_kernel<<<gA, 256, 0, stream>>>(in, scores, classes, boxes);
  topk_kernel<<<NB, 1024, 0, stream>>>(scores, classes, boxes, tk_s, tk_c, tk_b);
  nms_prep_kernel<<<NB, 1024, 0, stream>>>(tk_s, tk_c, tk_b, nms_b, areaw, validf);
  iou_mask_kernel<<<NB, 1024, 0, stream>>>(nms_b, areaw, maskw);
  nms_scan_kernel<<<NB, 1024, 0, stream>>>(tk_s, tk_c, tk_b, validf, maskw, out_sc, out_cl,
                                           out_bx, out_keep);
}